// GetStoneDistAngle3d_2422361555613
// MI455X (gfx1250) — compile-verified
//
#include <hip/hip_runtime.h>
#include <math.h>

#define S_STONES 128
#define THREADS 128
#define ROWS_PER_BLOCK 4

// ---- gfx1250 async global->LDS path (builtin confirmed present by round-1 probe) ----
#if defined(__gfx1250__) && __has_builtin(__builtin_amdgcn_global_load_async_to_lds_b32)
#define USE_ASYNC_LDS 1
#else
#define USE_ASYNC_LDS 0
#endif

typedef __attribute__((address_space(1))) int* gptr_i32;
typedef __attribute__((address_space(3))) int* lptr_i32;

__device__ __forceinline__ void wait_asynccnt0() {
#if __has_builtin(__builtin_amdgcn_s_wait_asynccnt)
  __builtin_amdgcn_s_wait_asynccnt(0);
#else
  asm volatile("s_wait_asynccnt 0" ::: "memory");
#endif
}

__global__ __launch_bounds__(THREADS) void stone_dist_angle_kernel(
    const float* __restrict__ coord,   // [N,3] (flag, y, x)
    const float* __restrict__ stone,   // [S,3] (value, y, x)
    float* __restrict__ out,           // [N,S,3]
    int n_rows)
{
  __shared__ float s_stone[3 * S_STONES];           // raw stone table
  __shared__ unsigned long long s_key[S_STONES];    // (dist_bits << 32) | idx
  __shared__ float s_out[3 * S_STONES];             // sorted row staging

  const int t = threadIdx.x;

  // Stage the 1.5KB stone table into LDS (coalesced, 3 b32 per thread).
#if USE_ASYNC_LDS
  {
#pragma unroll
    for (int c = 0; c < 3; ++c) {
      const int idx = t + c * THREADS;
      __builtin_amdgcn_global_load_async_to_lds_b32(
          (gptr_i32)(stone + idx),
          (lptr_i32)(&s_stone[idx]),
          0, 0);
    }
    wait_asynccnt0();
  }
#else
#pragma unroll
  for (int c = 0; c < 3; ++c) {
    const int idx = t + c * THREADS;
    s_stone[idx] = stone[idx];
  }
#endif
  __syncthreads();

  // This thread owns stone t for the whole block.
  const float sv = s_stone[3 * t + 0];
  const float sy = s_stone[3 * t + 1];
  const float sx = s_stone[3 * t + 2];

  const long long base_row = (long long)blockIdx.x * ROWS_PER_BLOCK;

  for (int r = 0; r < ROWS_PER_BLOCK; ++r) {
    const long long row = base_row + r;
    if (row >= n_rows) break;

    // Uniform per-block-row values (broadcast loads).
    const float flag = coord[row * 3 + 0];
    float* __restrict__ orow = out + row * (3 * S_STONES);

    if (flag == 0.0f) {
      const float cy = coord[row * 3 + 1];
      const float cx = coord[row * 3 + 2];

      const float dy = sy - cy;
      const float dx = sx - cx;
      const float dist = sqrtf(dy * dy + dx * dx);
      float ang = atan2f(-dy, dx) * 57.29577951308232087680f;  // 180/pi
      ang = (ang > 0.0f) ? ang : (360.0f + ang);

      // Unique, stable sort key: dist >= 0 so IEEE bits are monotone as u32;
      // low 32 bits carry the stone index for stable tiebreak (matches argsort).
      const unsigned long long key =
          ((unsigned long long)__float_as_uint(dist) << 32) | (unsigned)t;
      s_key[t] = key;
      __syncthreads();

      // Rank = number of strictly smaller keys (ranks form a perfect permutation).
      int rank = 0;
#pragma unroll 16
      for (int k = 0; k < S_STONES; ++k) {
        rank += (s_key[k] < key) ? 1 : 0;
      }

      // Scatter sorted triple into LDS, then coalesced copy-out.
      s_out[3 * rank + 0] = sv;
      s_out[3 * rank + 1] = dist;
      s_out[3 * rank + 2] = ang;
      __syncthreads();

      orow[t]               = s_out[t];
      orow[t + THREADS]     = s_out[t + THREADS];
      orow[t + 2 * THREADS] = s_out[t + 2 * THREADS];
    } else {
      // Flagged row -> zeros (coalesced).
      orow[t]               = 0.0f;
      orow[t + THREADS]     = 0.0f;
      orow[t + 2 * THREADS] = 0.0f;
    }
  }
}

extern "C" void kernel_launch(void* const* d_in, const int* in_sizes, int n_in,
                              void* d_out, int out_size, void* d_ws, size_t ws_size,
                              hipStream_t stream) {
  const float* coord = (const float*)d_in[0];  // [N,3] f32
  const float* stone = (const float*)d_in[1];  // [S,3] f32
  float* out = (float*)d_out;                  // [N,S,3] f32

  const int n_rows = in_sizes[0] / 3;          // N = 131072
  const int grid = (n_rows + ROWS_PER_BLOCK - 1) / ROWS_PER_BLOCK;

  stone_dist_angle_kernel<<<grid, THREADS, 0, stream>>>(coord, stone, out, n_rows);
}